// CGMMTransition_89172111000085
// MI455X (gfx1250) — compile-verified
//
#include <hip/hip_runtime.h>
#include <hip/hip_bf16.h>

// Problem constants
#define NN 10000
#define LL 5
#define AA 6
#define CC 20
#define CC2 20
#define K600 (LL*AA*CC2)        // 600: flattened (l,a,c2) contraction dim
#define LA (LL*AA)              // 30
#define ELEMS (NN*LL*AA*CC*CC2) // 120,000,000 per big output

typedef __attribute__((ext_vector_type(2))) float v2f;
typedef __attribute__((ext_vector_type(8))) float v8f;

// ---------------------------------------------------------------------------
// Kernel 1: tiny prep.
//   inv[n*30+la]  = 1 / sum_{c2} stats[n,la,c2]   (0 -> 1)
//   WT[c*600 + la*20 + c2] = layerS[l]*arcS[la]*transition[la,c,c2], c in [0,32) zero-padded
//   lw[la] = layerS[l]*arcS[la]
// ---------------------------------------------------------------------------
__global__ __launch_bounds__(256) void cgmm_prep(
    const float* __restrict__ stats, const float* __restrict__ layerS,
    const float* __restrict__ arcS, const float* __restrict__ transition,
    float* __restrict__ inv, float* __restrict__ WT, float* __restrict__ lw)
{
    int i = blockIdx.x * blockDim.x + threadIdx.x;
    if (i < NN * LA) {
        const float* p = &stats[(size_t)i * CC2];
        float s = 0.0f;
        #pragma unroll
        for (int j = 0; j < CC2; ++j) s += p[j];
        inv[i] = (s == 0.0f) ? 1.0f : 1.0f / s;
    }
    if (i < 32 * K600) {
        int c = i / K600;
        int r = i - c * K600;         // la*20 + c2
        int la = r / CC2;
        int c2 = r - la * CC2;
        float v = 0.0f;
        if (c < CC)
            v = layerS[la / AA] * arcS[la] *
                transition[(size_t)la * (CC * CC2) + c * CC2 + c2];
        WT[i] = v;
    }
    if (i < LA) lw[i] = layerS[i / AA] * arcS[i];
}

// ---------------------------------------------------------------------------
// Kernel 2: the bandwidth stream (960 MB of stores), float4-vectorized.
//   rightmost[n,la,c,c2]  = T[la,c,c2] * stats[n,la,c2] * inv[n,la]
//   posterior[n,la,c,c2]  = lw[la] * rightmost
// ---------------------------------------------------------------------------
__global__ __launch_bounds__(256) void cgmm_elemwise(
    const float* __restrict__ stats, const float* __restrict__ transition,
    const float* __restrict__ inv, const float* __restrict__ lw,
    float* __restrict__ posterior, float* __restrict__ rightmost)
{
    size_t q = (size_t)blockIdx.x * blockDim.x + threadIdx.x; // quad index
    const size_t totalQ = (size_t)ELEMS / 4;
    if (q >= totalQ) return;
    size_t e = q * 4;
    int n   = (int)(e / (LA * CC * CC2));          // e / 12000
    int rem = (int)(e - (size_t)n * (LA * CC * CC2));
    int la  = rem / (CC * CC2);                    // rem / 400
    int c2  = rem % CC2;                           // multiple of 4

    float4 T = *(const float4*)&transition[rem];   // rem == la*400 + c*20 + c2
    float4 s = *(const float4*)&stats[(size_t)n * K600 + la * CC2 + c2];
    float iv = inv[n * LA + la];
    float w  = lw[la];

    float4 r;
    r.x = T.x * s.x * iv;  r.y = T.y * s.y * iv;
    r.z = T.z * s.z * iv;  r.w = T.w * s.w * iv;
    *(float4*)&rightmost[e] = r;

    float4 p;
    p.x = r.x * w;  p.y = r.y * w;  p.z = r.z * w;  p.w = r.w * w;
    *(float4*)&posterior[e] = p;
}

// ---------------------------------------------------------------------------
// Kernel 3: p_Q via V_WMMA_F32_16X16X4_F32.
//   p_Q[n,c] = sum_k stats'[n,k] * WT[c,k],  k in [0,600)
// One wave32 per 16-row n-tile. A staged in LDS (already scaled by inv),
// B read from L2-resident WT (identical across all blocks).
// A frag (16x4 f32): lanes 0-15 hold M=lane, K=k0+{0,1}; lanes 16-31 M=lane-16, K=k0+{2,3}
// B frag (4x16 f32): lanes 0-15 hold N=lane, K=k0+{0,1}; lanes 16-31 N=lane-16, K=k0+{2,3}
// D (16x16 f32): VGPR v, lanes 0-15 -> M=v, N=lane; lanes 16-31 -> M=8+v, N=lane-16
// ---------------------------------------------------------------------------
__global__ __launch_bounds__(32) void cgmm_pq_wmma(
    const float* __restrict__ stats, const float* __restrict__ inv,
    const float* __restrict__ WT, float* __restrict__ pq)
{
    __shared__ float sInv[16 * LA];    // 480 floats
    __shared__ float sA[16 * K600];    // 9600 floats (38.4 KB)

    const int lane = threadIdx.x;
    const int n0   = blockIdx.x * 16;

    // stage inv for this 16-row tile (480 = 15*32, no divergence)
    #pragma unroll
    for (int i = lane; i < 16 * LA; i += 32) {
        int m = i / LA;
        sInv[i] = inv[(n0 + m) * LA + (i - m * LA)];
    }
    __syncthreads();

    // stage scaled stats' = stats * inv  (9600 = 300*32, no divergence)
    for (int i = lane; i < 16 * K600; i += 32) {
        int m = i / K600;
        int k = i - m * K600;
        sA[i] = stats[(size_t)(n0 + m) * K600 + k] * sInv[m * LA + k / CC2];
    }
    __syncthreads();

    const int col = lane & 15;          // M for A-frag, N (c) for B-frag
    const int kb  = (lane >> 4) << 1;   // lane-half K offset {0,2}
    const float* Arow = &sA[col * K600];
    const float* Brow0 = &WT[col * K600];
    const float* Brow1 = &WT[(col + 16) * K600];

    v8f acc0 = {0.f,0.f,0.f,0.f,0.f,0.f,0.f,0.f};
    v8f acc1 = {0.f,0.f,0.f,0.f,0.f,0.f,0.f,0.f};

    #pragma unroll 4
    for (int k0 = 0; k0 < K600; k0 += 4) {
        v2f a  = *(const v2f*)&Arow [k0 + kb];   // LDS, 8B aligned
        v2f b0 = *(const v2f*)&Brow0[k0 + kb];   // global (L2-hot), 8B aligned
        v2f b1 = *(const v2f*)&Brow1[k0 + kb];
        acc0 = __builtin_amdgcn_wmma_f32_16x16x4_f32(
            false, a, false, b0, (short)0, acc0, false, false);
        acc1 = __builtin_amdgcn_wmma_f32_16x16x4_f32(
            false, a, false, b1, (short)0, acc1, false, false);
    }

    // scatter the 16x20 result tile
    const int mbase = (lane >> 4) * 8;
    #pragma unroll
    for (int v = 0; v < 8; ++v) {
        int m = mbase + v;
        pq[(size_t)(n0 + m) * CC + col] = acc0[v];
        if (col < CC - 16)
            pq[(size_t)(n0 + m) * CC + 16 + col] = acc1[v];
    }
}

// ---------------------------------------------------------------------------
extern "C" void kernel_launch(void* const* d_in, const int* in_sizes, int n_in,
                              void* d_out, int out_size, void* d_ws, size_t ws_size,
                              hipStream_t stream) {
    const float* stats      = (const float*)d_in[0];
    const float* layerS     = (const float*)d_in[1];
    const float* arcS       = (const float*)d_in[2];
    const float* transition = (const float*)d_in[3];

    float* out       = (float*)d_out;
    float* pq        = out;                         // [N*C]           = 200,000
    float* posterior = out + (size_t)NN * CC;       // [N,L,A,C,C2]    = 120,000,000
    float* rightmost = posterior + (size_t)ELEMS;   // [N,L,A,C,C2]    = 120,000,000

    float* inv = (float*)d_ws;                      // 300,000 floats
    float* WT  = inv + (size_t)NN * LA;             // 32*600 = 19,200 floats (zero-padded)
    float* lw  = WT + 32 * K600;                    // 30 floats

    // 1) prep (covers max(300000, 19200, 30) threads)
    {
        int threads = NN * LA;                      // 300,000
        int blocks  = (threads + 255) / 256;
        cgmm_prep<<<blocks, 256, 0, stream>>>(stats, layerS, arcS, transition,
                                              inv, WT, lw);
    }
    // 2) elementwise stream: 30,000,000 float4 quads
    {
        int quads  = ELEMS / 4;
        int blocks = (quads + 255) / 256;
        cgmm_elemwise<<<blocks, 256, 0, stream>>>(stats, transition, inv, lw,
                                                  posterior, rightmost);
    }
    // 3) p_Q GEMM via WMMA: 625 tiles of 16 rows, one wave each
    {
        cgmm_pq_wmma<<<NN / 16, 32, 0, stream>>>(stats, inv, WT, pq);
    }
}